// SpatialAttentionModule_67680094650428
// MI455X (gfx1250) — compile-verified
//
#include <hip/hip_runtime.h>
#include <hip/hip_bf16.h>
#include <math.h>

#define BATCH  8
#define NNODES 2048
#define HIDDEN 128
#define HEADS  4
#define HDIM   32
#define QKV3   384   // 3*HIDDEN

typedef __attribute__((ext_vector_type(2))) float v2f;
typedef __attribute__((ext_vector_type(8))) float v8f;

// Branchless elu(x)+1 = exp(min(x,0)) + max(x,0).
// Argument of exp is <= 0 -> no overflow path, v_exp_f32 is safe and exact
// enough (f32 hardware exp2 with ~1 ulp), no EXEC-diverging branches.
__device__ __forceinline__ float elu1(float x) {
  return __expf(fminf(x, 0.0f)) + fmaxf(x, 0.0f);
}

// ---------------------------------------------------------------------------
// Fold low-rank weight pairs:  W_comb = W_ql @ W_qh,  b_comb = b_ql@W_qh + b_qh
// ---------------------------------------------------------------------------
__global__ void combine_qkv_w_kernel(const float* __restrict__ W_ql,
                                     const float* __restrict__ b_ql,
                                     const float* __restrict__ W_qh,
                                     const float* __restrict__ b_qh,
                                     float* __restrict__ W_comb,
                                     float* __restrict__ b_comb) {
  int idx = blockIdx.x * blockDim.x + threadIdx.x;
  if (idx < HIDDEN * QKV3) {
    int i = idx / QKV3, j = idx % QKV3;
    float s = 0.0f;
    #pragma unroll
    for (int r = 0; r < 24; ++r) s = fmaf(W_ql[i * 24 + r], W_qh[r * QKV3 + j], s);
    W_comb[idx] = s;
  }
  if (idx < QKV3) {
    float s = b_qh[idx];
    #pragma unroll
    for (int r = 0; r < 24; ++r) s = fmaf(b_ql[r], W_qh[r * QKV3 + idx], s);
    b_comb[idx] = s;
  }
}

__global__ void combine_out_w_kernel(const float* __restrict__ W_ol,
                                     const float* __restrict__ b_ol,
                                     const float* __restrict__ W_oh,
                                     const float* __restrict__ b_oh,
                                     float* __restrict__ Wo_comb,
                                     float* __restrict__ bo_comb) {
  int idx = blockIdx.x * blockDim.x + threadIdx.x;
  if (idx < HIDDEN * HIDDEN) {
    int i = idx / HIDDEN, j = idx % HIDDEN;
    float s = 0.0f;
    #pragma unroll
    for (int r = 0; r < 8; ++r) s = fmaf(W_ol[i * 8 + r], W_oh[r * HIDDEN + j], s);
    Wo_comb[idx] = s;
  }
  if (idx < HIDDEN) {
    float s = b_oh[idx];
    #pragma unroll
    for (int r = 0; r < 8; ++r) s = fmaf(b_ol[r], W_oh[r * HIDDEN + idx], s);
    bo_comb[idx] = s;
  }
}

// ---------------------------------------------------------------------------
// C[M,N] = A[M,K] @ B[K,N] + bias[N], row-major f32, V_WMMA_F32_16X16X4_F32.
// N, K compile-time -> all inner-loop addresses become immediate offsets.
// blockDim.x == 256 (8 waves). Each wave computes a 16 x (16*NT) tile,
// reusing one A fragment across NT WMMAs per k-step.
// B tile is staged ONCE per block into LDS in a paired-k layout
// Bs[k/2][col][2] (pair-row stride padded to STR floats so the two EXEC
// halves of a wave hit disjoint bank halves); one ds_load_b64 then yields
// both k-rows of a B fragment.
// grid: x = M/128, y = N/(16*NT).
// A frag: lane L -> A[m0 + L%16, k0 + (L/16)*2 + j]   (j = vgpr 0/1)
// B frag: lane L -> B[k0 + (L/16)*2 + j, n0 + L%16]
// D:      vgpr r -> C[m0 + (L/16)*8 + r, n0 + L%16]
// ---------------------------------------------------------------------------
template<int N, int K, int NT>
__global__ void wmma_gemm_f32_tiled(const float* __restrict__ A,
                                    const float* __restrict__ B,
                                    const float* __restrict__ bias,
                                    float* __restrict__ C) {
  constexpr int NCOL = 16 * NT;            // 64 output cols per block
  constexpr int STR  = NCOL * 2 + 32;      // padded pair-row stride (floats)
  __shared__ float Bs[(K / 2) * STR];      // 40 KB for K=128

  const int tid   = threadIdx.x;
  const int lane  = tid & 31;
  const int wave  = tid >> 5;
  const int m0    = (blockIdx.x * 8 + wave) * 16;
  const int n0    = blockIdx.y * NCOL;
  const int row   = lane & 15;
  const int khalf = (lane >> 4) * 2;

  // ---- stage B tile [K][NCOL] -> paired LDS layout (b128 global loads) ----
  constexpr int UNITS = (K / 2) * (NCOL / 4);   // 4-col units per k-pair
  #pragma unroll
  for (int u = tid; u < UNITS; u += 256) {
    int p = u / (NCOL / 4);                     // k-pair index
    int c = (u % (NCOL / 4)) * 4;               // col
    const float* g0 = B + (size_t)(2 * p) * N + n0 + c;
    float4 r0 = *(const float4*)g0;             // row 2p,   cols c..c+3
    float4 r1 = *(const float4*)(g0 + N);       // row 2p+1, cols c..c+3
    float* dst = &Bs[p * STR + c * 2];
    ((v2f*)dst)[0] = (v2f){r0.x, r1.x};
    ((v2f*)dst)[1] = (v2f){r0.y, r1.y};
    ((v2f*)dst)[2] = (v2f){r0.z, r1.z};
    ((v2f*)dst)[3] = (v2f){r0.w, r1.w};
  }
  __syncthreads();

  const float* Arow = A + (size_t)(m0 + row) * K + khalf;   // + k0 (imm)

  v8f acc[NT];
  #pragma unroll
  for (int t = 0; t < NT; ++t) {
    v8f zero = {};
    acc[t] = zero;
  }

  #pragma unroll
  for (int k0 = 0; k0 < K; k0 += 4) {
    v2f a = *(const v2f*)(Arow + k0);                         // 8B global load
    const float* bp = &Bs[(k0 / 2 + (lane >> 4)) * STR + row * 2];
    #pragma unroll
    for (int t = 0; t < NT; ++t) {
      v2f b = *(const v2f*)(bp + 32 * t);                     // ds_load_b64
      acc[t] = __builtin_amdgcn_wmma_f32_16x16x4_f32(false, a, false, b,
                                                     (short)0, acc[t],
                                                     false, false);
    }
  }

  const int mbase = m0 + (lane >> 4) * 8;
  #pragma unroll
  for (int t = 0; t < NT; ++t) {
    const int   col = n0 + 16 * t + row;
    const float bb  = bias[col];
    #pragma unroll
    for (int r = 0; r < 8; ++r)
      C[(size_t)(mbase + r) * N + col] = acc[t][r] + bb;
  }
}

// ---------------------------------------------------------------------------
// Per (b,h): kv[d][e] = sum_n elu1(k[n,d]) * v[n,e]   (32x32 reduction over 2048)
//            z[n]     = 1 / (sum_d elu1(k[n,d]) + 1e-8)
// One block per (b,h); LDS-chunked f32 FMA reduction (full f32 accuracy).
// ---------------------------------------------------------------------------
__global__ void kv_z_kernel(const float* __restrict__ qkv,
                            float* __restrict__ kv,
                            float* __restrict__ z) {
  const int bh  = blockIdx.x;           // 0..31
  const int b   = bh >> 2, h = bh & 3;
  const int tid = threadIdx.x;          // 0..255

  __shared__ float kfs[64][33];
  __shared__ float vs[64][33];

  const int d  = tid >> 3;              // 0..31
  const int e0 = (tid & 7) * 4;         // 0,4,...,28
  float acc0 = 0.f, acc1 = 0.f, acc2 = 0.f, acc3 = 0.f;

  const size_t base = (size_t)b * NNODES * QKV3 + (size_t)h * HDIM;

  for (int c = 0; c < NNODES; c += 64) {
    for (int idx = tid; idx < 64 * HDIM; idx += 256) {
      int r  = idx >> 5;
      int dd = idx & 31;
      size_t off = base + (size_t)(c + r) * QKV3 + dd;
      kfs[r][dd] = elu1(qkv[off + HIDDEN]);      // k slice -> feature map
      vs[r][dd]  = qkv[off + 2 * HIDDEN];        // v slice
    }
    __syncthreads();
    if (tid < 64) {
      float s = 0.f;
      #pragma unroll
      for (int dd = 0; dd < HDIM; ++dd) s += kfs[tid][dd];
      z[(size_t)bh * NNODES + c + tid] = 1.0f / (s + 1e-8f);
    }
    #pragma unroll 4
    for (int r = 0; r < 64; ++r) {
      float kf = kfs[r][d];
      acc0 = fmaf(kf, vs[r][e0 + 0], acc0);
      acc1 = fmaf(kf, vs[r][e0 + 1], acc1);
      acc2 = fmaf(kf, vs[r][e0 + 2], acc2);
      acc3 = fmaf(kf, vs[r][e0 + 3], acc3);
    }
    __syncthreads();
  }

  float* kvbh = kv + (size_t)bh * (HDIM * HDIM);
  kvbh[d * HDIM + e0 + 0] = acc0;
  kvbh[d * HDIM + e0 + 1] = acc1;
  kvbh[d * HDIM + e0 + 2] = acc2;
  kvbh[d * HDIM + e0 + 3] = acc3;
}

// ---------------------------------------------------------------------------
// Per (b,h): lin[n,e] = z[n] * sum_d elu1(q[n,d]) * kv[d,e], via f32 WMMA.
// One wave computes one 16-node tile for BOTH 16-wide head-dim halves
// (reuses the elu-transformed A fragment). Stores lin as [B, N, HIDDEN].
// grid: x = NNODES/128, y = 32 (bh).  blockDim = 256 (8 waves).
// ---------------------------------------------------------------------------
__global__ void lin_wmma_kernel(const float* __restrict__ qkv,
                                const float* __restrict__ kv,
                                const float* __restrict__ z,
                                float* __restrict__ lin) {
  const int lane  = threadIdx.x & 31;
  const int wave  = threadIdx.x >> 5;
  const int bh    = blockIdx.y;
  const int b     = bh >> 2, h = bh & 3;
  const int m0    = (blockIdx.x * 8 + wave) * 16;   // node tile
  const int row   = lane & 15;
  const int khalf = (lane >> 4) * 2;

  const float* qrow = qkv + ((size_t)b * NNODES + (m0 + row)) * QKV3 + h * HDIM + khalf;
  const float* kvb  = kv + (size_t)bh * (HDIM * HDIM) + (size_t)khalf * HDIM + row;

  v8f acc0 = {}, acc1 = {};
  #pragma unroll
  for (int k0 = 0; k0 < HDIM; k0 += 4) {
    v2f a;
    a.x = elu1(qrow[k0 + 0]);
    a.y = elu1(qrow[k0 + 1]);
    v2f b0, b1;
    b0.x = kvb[k0 * HDIM];                 // e-tile 0, row k0+khalf
    b0.y = kvb[k0 * HDIM + HDIM];          // e-tile 0, row k0+khalf+1
    b1.x = kvb[k0 * HDIM + 16];            // e-tile 1
    b1.y = kvb[k0 * HDIM + HDIM + 16];
    acc0 = __builtin_amdgcn_wmma_f32_16x16x4_f32(false, a, false, b0,
                                                 (short)0, acc0, false, false);
    acc1 = __builtin_amdgcn_wmma_f32_16x16x4_f32(false, a, false, b1,
                                                 (short)0, acc1, false, false);
  }

  const int    mbase = m0 + (lane >> 4) * 8;
  const float* zb    = z + (size_t)bh * NNODES;
  #pragma unroll
  for (int r = 0; r < 8; ++r) {
    int    n  = mbase + r;
    float  zn = zb[n];
    float* lr = lin + ((size_t)b * NNODES + n) * HIDDEN + h * HDIM + row;
    lr[0]  = acc0[r] * zn;
    lr[16] = acc1[r] * zn;
  }
}

// ---------------------------------------------------------------------------
// reg_loss = exp(log_reg) * mean(|softmax(.)|) = exp(log_reg) * (1/N) exactly,
// since softmax rows are non-negative and sum to 1.
// ---------------------------------------------------------------------------
__global__ void reg_kernel(const float* __restrict__ log_reg,
                           float* __restrict__ out_reg) {
  if (threadIdx.x == 0 && blockIdx.x == 0)
    out_reg[0] = expf(log_reg[0]) * (1.0f / (float)NNODES);
}

extern "C" void kernel_launch(void* const* d_in, const int* in_sizes, int n_in,
                              void* d_out, int out_size, void* d_ws, size_t ws_size,
                              hipStream_t stream) {
  (void)in_sizes; (void)n_in; (void)out_size; (void)ws_size;
  const float* x       = (const float*)d_in[0];
  const float* W_ql    = (const float*)d_in[1];
  const float* b_ql    = (const float*)d_in[2];
  const float* W_qh    = (const float*)d_in[3];
  const float* b_qh    = (const float*)d_in[4];
  const float* W_ol    = (const float*)d_in[5];
  const float* b_ol    = (const float*)d_in[6];
  const float* W_oh    = (const float*)d_in[7];
  const float* b_oh    = (const float*)d_in[8];
  // d_in[9]=u, d_in[10]=v: dead inputs (only feed the constant reg_loss term)
  const float* log_reg = (const float*)d_in[11];
  float* out = (float*)d_out;

  float* ws      = (float*)d_ws;
  float* W_comb  = ws;                                     // 128*384
  float* b_comb  = W_comb + HIDDEN * QKV3;                 // 384
  float* Wo_comb = b_comb + QKV3;                          // 128*128
  float* bo_comb = Wo_comb + HIDDEN * HIDDEN;              // 128
  float* qkv     = bo_comb + HIDDEN;                       // 8*2048*384
  float* kvbuf   = qkv + (size_t)BATCH * NNODES * QKV3;    // 32*1024
  float* zbuf    = kvbuf + 32 * HDIM * HDIM;               // 32*2048
  float* lin     = zbuf + 32 * NNODES;                     // 8*2048*128

  const int MROWS = BATCH * NNODES;   // 16384

  combine_qkv_w_kernel<<<(HIDDEN * QKV3 + 255) / 256, 256, 0, stream>>>(
      W_ql, b_ql, W_qh, b_qh, W_comb, b_comb);
  combine_out_w_kernel<<<(HIDDEN * HIDDEN + 255) / 256, 256, 0, stream>>>(
      W_ol, b_ol, W_oh, b_oh, Wo_comb, bo_comb);

  // qkv = x @ W_comb + b_comb     [16384,128]@[128,384], 16x64 wave tiles
  wmma_gemm_f32_tiled<QKV3, HIDDEN, 4>
      <<<dim3(MROWS / 128, QKV3 / 64), 256, 0, stream>>>(x, W_comb, b_comb, qkv);

  // per-(b,h) kv and z
  kv_z_kernel<<<32, 256, 0, stream>>>(qkv, kvbuf, zbuf);

  // lin = (elu(q)+1) @ kv * z     per (b,h): [2048,32]@[32,32]
  lin_wmma_kernel<<<dim3(NNODES / 128, 32), 256, 0, stream>>>(
      qkv, kvbuf, zbuf, lin);

  // out = lin @ Wo_comb + bo_comb [16384,128]@[128,128], 16x64 wave tiles
  wmma_gemm_f32_tiled<HIDDEN, HIDDEN, 4>
      <<<dim3(MROWS / 128, HIDDEN / 64), 256, 0, stream>>>(lin, Wo_comb, bo_comb, out);

  // reg_loss scalar appended after the [8,2048,128] output
  reg_kernel<<<1, 32, 0, stream>>>(log_reg, out + (size_t)MROWS * HIDDEN);
}